// STGCNBlock_1992864825921
// MI455X (gfx1250) — compile-verified
//
#include <hip/hip_runtime.h>

typedef float v2f __attribute__((ext_vector_type(2)));
typedef float v8f __attribute__((ext_vector_type(8)));

#define T_DIM   32
#define N_NODES 10000
#define CIN     32
#define COUT    64
#define N_EDGES 160000
#define NROWS   (T_DIM * N_NODES)   // 320000
#define EPS_BN  1e-5f

__device__ __forceinline__ v8f wmma4(v2f a, v2f b, v8f c) {
  // D = A(16x4 f32) * B(4x16 f32) + C(16x16 f32)
  return __builtin_amdgcn_wmma_f32_16x16x4_f32(false, a, false, b, (short)0, c, false, false);
}

// packed-B index: row r (K dim), col c -> fragment-pair layout so each lane
// loads its (B[kb][col], B[kb+1][col]) pair as one aligned b64.
__device__ __forceinline__ int packB(int r, int c) {
  return (((r >> 2) * 2 + ((r >> 1) & 1)) << 7) + (c << 1) + (r & 1);
}

// ---------------- degree / dinv ----------------
__global__ void deg_kernel(const int* __restrict__ ei, float* __restrict__ deg) {
  int e = blockIdx.x * blockDim.x + threadIdx.x;
  if (e >= N_EDGES) return;
  atomicAdd(&deg[ei[N_EDGES + e]], 1.0f);   // dst list
}

__global__ void dinv_kernel(float* __restrict__ deg) {
  int n = blockIdx.x * blockDim.x + threadIdx.x;
  if (n >= N_NODES) return;
  deg[n] = rsqrtf(deg[n] + 1.0f);           // +1 self loop; always > 0
}

// ---------------- weight reformat into WMMA fragment-pair order ----------------
// Wp  [8 ksteps *2*64*2]            from gcn_w  [32,64]
// Bwp [3][16 ksteps *2*64*2]        from tconv_w[o][i][k]  (B = W_k^T, [64,64])
// Brp [8 ksteps *2*64*2]            from res_w  [64,32]    (B = res_w^T, [32,64])
__global__ void prep_weights(const float* __restrict__ gw, const float* __restrict__ tw,
                             const float* __restrict__ rw, float* __restrict__ Wp,
                             float* __restrict__ Bwp, float* __restrict__ Brp) {
  int tid = blockIdx.x * blockDim.x + threadIdx.x;
  if (tid < CIN * COUT) {                       // gcn_w: r=f (0..31), c (0..63)
    int r = tid >> 6, c = tid & 63;
    Wp[packB(r, c)] = gw[tid];
  } else if (tid < CIN * COUT + 3 * COUT * COUT) {   // tconv: B_k[i][o] = tw[o][i][k]
    int q = tid - CIN * COUT;
    int k = q / (COUT * COUT);
    int rem = q - k * (COUT * COUT);
    int i = rem >> 6, o = rem & 63;
    Bwp[k * (COUT * COUT) + packB(i, o)] = tw[(o * COUT + i) * 3 + k];
  } else if (tid < CIN * COUT + 3 * COUT * COUT + CIN * COUT) {  // res: B[i][o] = rw[o][i]
    int q = tid - CIN * COUT - 3 * COUT * COUT;
    int i = q >> 6, o = q & 63;
    Brp[packB(i, o)] = rw[o * CIN + i];
  }
}

// ---------------- GEMM1: h = x @ gcn_w  [NROWS,32]x[32,64] ----------------
__global__ __launch_bounds__(256) void gemm_gcn(const float* __restrict__ X,
                                                const float* __restrict__ Wp,
                                                float* __restrict__ H) {
  const int lane = threadIdx.x & 31;
  const int l15 = lane & 15, hi = lane >> 4;
  const int rowTile = blockIdx.x * 8 + (threadIdx.x >> 5);
  const int row0 = rowTile * 16;
  v8f a0 = {}, a1 = {}, a2 = {}, a3 = {};
  const float* arow = X + (size_t)(row0 + l15) * CIN;
#pragma unroll
  for (int kk = 0; kk < 8; ++kk) {
    const int kb = kk * 4 + hi * 2;
    v2f a = *(const v2f*)(arow + kb);
    const float* bp = Wp + ((kk * 2 + hi) << 7);
    a0 = wmma4(a, *(const v2f*)(bp + (l15 << 1)), a0);
    a1 = wmma4(a, *(const v2f*)(bp + ((16 + l15) << 1)), a1);
    a2 = wmma4(a, *(const v2f*)(bp + ((32 + l15) << 1)), a2);
    a3 = wmma4(a, *(const v2f*)(bp + ((48 + l15) << 1)), a3);
  }
#pragma unroll
  for (int r = 0; r < 8; ++r) {
    float* o = H + (size_t)(row0 + r + hi * 8) * COUT + l15;
    o[0] = a0[r]; o[16] = a1[r]; o[32] = a2[r]; o[48] = a3[r];
  }
}

// ---------------- edge scatter: agg[t,dst,:] += h[t,src,:]*dinv[s]*dinv[d] ----------------
__global__ void edge_agg(const int* __restrict__ ei, const float* __restrict__ dinv,
                         const float* __restrict__ H, float* __restrict__ AGG) {
  long long gid = (long long)blockIdx.x * blockDim.x + threadIdx.x;
  if (gid >= (long long)N_EDGES * T_DIM) return;
  const int e = (int)(gid >> 5);
  const int t = (int)(gid & 31);
  const int s = ei[e];
  const int d = ei[N_EDGES + e];
  const float w = dinv[s] * dinv[d];
  const float* hr = H + ((size_t)t * N_NODES + s) * COUT;
  float* ar = AGG + ((size_t)t * N_NODES + d) * COUT;
#pragma unroll
  for (int c = 0; c < COUT; c += 4) {
    const float4 hv = *(const float4*)(hr + c);
    atomicAdd(ar + c + 0, hv.x * w);
    atomicAdd(ar + c + 1, hv.y * w);
    atomicAdd(ar + c + 2, hv.z * w);
    atomicAdd(ar + c + 3, hv.w * w);
  }
}

// self loops: atomic-free (runs after edge_agg on same stream)
__global__ void self_agg(const float* __restrict__ dinv, const float* __restrict__ H,
                         float* __restrict__ AGG) {
  int gid = blockIdx.x * blockDim.x + threadIdx.x;
  if (gid >= NROWS) return;
  const int n = gid % N_NODES;
  const float w = dinv[n] * dinv[n];
  const float* hr = H + (size_t)gid * COUT;
  float* ar = AGG + (size_t)gid * COUT;
#pragma unroll
  for (int c = 0; c < COUT; c += 4) {
    float4 a = *(const float4*)(ar + c);
    const float4 hv = *(const float4*)(hr + c);
    a.x += hv.x * w; a.y += hv.y * w; a.z += hv.z * w; a.w += hv.w * w;
    *(float4*)(ar + c) = a;
  }
}

// ---------------- BN per timestep over N (stats) ----------------
__global__ __launch_bounds__(256) void bns_stats_k(const float* __restrict__ AGG,
                                                   float* __restrict__ stats) {
  const int t = blockIdx.x;        // 0..31
  const int chunk = blockIdx.y;    // 0..15
  const int c = threadIdx.x & 63, r = threadIdx.x >> 6;
  const float* base = AGG + (size_t)t * N_NODES * COUT;
  const int n0 = chunk * 625;
  float s = 0.f, s2 = 0.f;
  for (int n = n0 + r; n < n0 + 625; n += 4) {
    float v = base[(size_t)n * COUT + c];
    s += v; s2 += v * v;
  }
  __shared__ float ls[64], ls2[64];
  if (r == 0) { ls[c] = 0.f; ls2[c] = 0.f; }
  __syncthreads();
  atomicAdd(&ls[c], s);
  atomicAdd(&ls2[c], s2);
  __syncthreads();
  if (threadIdx.x < 64) {
    atomicAdd(&stats[t * 64 + threadIdx.x], ls[threadIdx.x]);
    atomicAdd(&stats[2048 + t * 64 + threadIdx.x], ls2[threadIdx.x]);
  }
}

__global__ void bns_fin(float* __restrict__ stats, const float* __restrict__ gamma,
                        const float* __restrict__ beta) {
  int tid = blockIdx.x * blockDim.x + threadIdx.x;
  if (tid >= 2048) return;
  const int c = tid & 63;
  float s = stats[tid], s2 = stats[2048 + tid];
  float m = s / (float)N_NODES;
  float var = s2 / (float)N_NODES - m * m;
  float sc = gamma[c] * rsqrtf(var + EPS_BN);
  stats[tid] = sc;                      // scale
  stats[2048 + tid] = beta[c] - m * sc; // shift
}

__global__ void bns_apply(float* __restrict__ AGG, const float* __restrict__ stats) {
  long long gid = (long long)blockIdx.x * blockDim.x + threadIdx.x;
  if (gid >= (long long)NROWS * 16) return;
  const size_t o = (size_t)gid * 4;
  const int t = (int)(o / ((size_t)N_NODES * COUT));
  const int c = (int)(o & 63);
  const float* sc = stats + t * 64;
  const float* sh = stats + 2048 + t * 64;
  float4 v = *(float4*)(AGG + o);
  v.x = fmaxf(0.f, v.x * sc[c + 0] + sh[c + 0]);
  v.y = fmaxf(0.f, v.y * sc[c + 1] + sh[c + 1]);
  v.z = fmaxf(0.f, v.z * sc[c + 2] + sh[c + 2]);
  v.w = fmaxf(0.f, v.w * sc[c + 3] + sh[c + 3]);
  *(float4*)(AGG + o) = v;
}

// ---------------- temporal conv: 3 shifted [.,64]x[64,64] GEMMs ----------------
__global__ __launch_bounds__(256) void tconv_gemm(const float* __restrict__ HS,
                                                  const float* __restrict__ BWp,
                                                  float* __restrict__ OUT) {
  const int lane = threadIdx.x & 31;
  const int l15 = lane & 15, hi = lane >> 4;
  const int rowTile = blockIdx.x * 8 + (threadIdx.x >> 5);
  const int t = rowTile / 625;
  const int n0 = (rowTile % 625) * 16;
  v8f a0 = {}, a1 = {}, a2 = {}, a3 = {};
#pragma unroll
  for (int k = 0; k < 3; ++k) {
    const int ts = t + k - 1;                 // wave-uniform predicate: EXEC stays full
    if (ts < 0 || ts >= T_DIM) continue;
    const float* arow = HS + ((size_t)ts * N_NODES + n0 + l15) * COUT;
    const float* Bk = BWp + k * (COUT * COUT);
#pragma unroll
    for (int kk = 0; kk < 16; ++kk) {
      const int kb = kk * 4 + hi * 2;
      v2f a = *(const v2f*)(arow + kb);
      const float* bp = Bk + ((kk * 2 + hi) << 7);
      a0 = wmma4(a, *(const v2f*)(bp + (l15 << 1)), a0);
      a1 = wmma4(a, *(const v2f*)(bp + ((16 + l15) << 1)), a1);
      a2 = wmma4(a, *(const v2f*)(bp + ((32 + l15) << 1)), a2);
      a3 = wmma4(a, *(const v2f*)(bp + ((48 + l15) << 1)), a3);
    }
  }
  const int row0 = t * N_NODES + n0;
#pragma unroll
  for (int r = 0; r < 8; ++r) {
    float* o = OUT + (size_t)(row0 + r + hi * 8) * COUT + l15;
    o[0] = a0[r]; o[16] = a1[r]; o[32] = a2[r]; o[48] = a3[r];
  }
}

// ---------------- BN over all rows (stats) ----------------
__global__ __launch_bounds__(256) void bnt_stats_k(const float* __restrict__ OUT,
                                                   float* __restrict__ accum) {
  const int c = threadIdx.x & 63, r = threadIdx.x >> 6;
  const int r0 = blockIdx.x * 625;
  float s = 0.f, s2 = 0.f;
  for (int row = r0 + r; row < r0 + 625; row += 4) {
    float v = OUT[(size_t)row * COUT + c];
    s += v; s2 += v * v;
  }
  __shared__ float ls[64], ls2[64];
  if (r == 0) { ls[c] = 0.f; ls2[c] = 0.f; }
  __syncthreads();
  atomicAdd(&ls[c], s);
  atomicAdd(&ls2[c], s2);
  __syncthreads();
  if (threadIdx.x < 64) {
    atomicAdd(&accum[threadIdx.x], ls[threadIdx.x]);
    atomicAdd(&accum[64 + threadIdx.x], ls2[threadIdx.x]);
  }
}

__global__ void bnt_fin(float* __restrict__ accum, const float* __restrict__ gamma,
                        const float* __restrict__ beta) {
  int c = threadIdx.x;
  if (c >= 64) return;
  const float M = (float)NROWS;
  float m = accum[c] / M;
  float var = accum[64 + c] / M - m * m;
  float sc = gamma[c] * rsqrtf(var + EPS_BN);
  accum[128 + c] = sc;
  accum[192 + c] = beta[c] - m * sc;
}

// ---------------- final: y = relu(BN(ht)) + x @ res_w^T + res_b (in-place on OUT) ----------------
__global__ __launch_bounds__(256) void final_res(const float* __restrict__ X,
                                                 const float* __restrict__ Brp,
                                                 const float* __restrict__ accum,
                                                 const float* __restrict__ resb,
                                                 float* __restrict__ OUT) {
  const int lane = threadIdx.x & 31;
  const int l15 = lane & 15, hi = lane >> 4;
  const int rowTile = blockIdx.x * 8 + (threadIdx.x >> 5);
  const int row0 = rowTile * 16;
  v8f a0 = {}, a1 = {}, a2 = {}, a3 = {};
  const float* arow = X + (size_t)(row0 + l15) * CIN;
#pragma unroll
  for (int kk = 0; kk < 8; ++kk) {
    const int kb = kk * 4 + hi * 2;
    v2f a = *(const v2f*)(arow + kb);
    const float* bp = Brp + ((kk * 2 + hi) << 7);
    a0 = wmma4(a, *(const v2f*)(bp + (l15 << 1)), a0);
    a1 = wmma4(a, *(const v2f*)(bp + ((16 + l15) << 1)), a1);
    a2 = wmma4(a, *(const v2f*)(bp + ((32 + l15) << 1)), a2);
    a3 = wmma4(a, *(const v2f*)(bp + ((48 + l15) << 1)), a3);
  }
  const float* sc = accum + 128;
  const float* sh = accum + 192;
#pragma unroll
  for (int r = 0; r < 8; ++r) {
    float* o = OUT + (size_t)(row0 + r + hi * 8) * COUT + l15;
    o[0]  = fmaxf(0.f, o[0]  * sc[l15]      + sh[l15])      + a0[r] + resb[l15];
    o[16] = fmaxf(0.f, o[16] * sc[l15 + 16] + sh[l15 + 16]) + a1[r] + resb[l15 + 16];
    o[32] = fmaxf(0.f, o[32] * sc[l15 + 32] + sh[l15 + 32]) + a2[r] + resb[l15 + 32];
    o[48] = fmaxf(0.f, o[48] * sc[l15 + 48] + sh[l15 + 48]) + a3[r] + resb[l15 + 48];
  }
}

extern "C" void kernel_launch(void* const* d_in, const int* in_sizes, int n_in,
                              void* d_out, int out_size, void* d_ws, size_t ws_size,
                              hipStream_t stream) {
  const float* x       = (const float*)d_in[0];
  const int*   ei      = (const int*)d_in[1];
  const float* gcn_w   = (const float*)d_in[2];
  // d_in[3] gcn_b: cancelled by BN_s
  const float* bns_g   = (const float*)d_in[4];
  const float* bns_b   = (const float*)d_in[5];
  const float* tconv_w = (const float*)d_in[6];
  // d_in[7] tconv_b: cancelled by BN_t
  const float* bnt_g   = (const float*)d_in[8];
  const float* bnt_b   = (const float*)d_in[9];
  const float* res_w   = (const float*)d_in[10];
  const float* res_b   = (const float*)d_in[11];
  float* out = (float*)d_out;

  // workspace layout (float offsets)
  float* ws    = (float*)d_ws;
  float* dinv  = ws;                 // N floats (deg -> dinv in place)
  float* stats = ws + 16384;         // 4096 floats: per-(t,c) sum/sumsq -> scale/shift
  float* accum = ws + 20480;         // 256 floats: global sum/sumsq/scale/shift
  float* Bwp   = ws + 20736;         // 3*64*64 = 12288 floats (packed)
  float* Brp   = ws + 33024;         // 32*64  = 2048 floats (packed)
  float* Wp    = ws + 35072;         // 32*64  = 2048 floats (packed)
  float* H     = ws + 65536;         // [T,N,64] = 20,480,000 floats
  float* AGG   = H + (size_t)NROWS * COUT;  // [T,N,64] (becomes hs in place)

  // zero accumulators (graph-capture safe)
  hipMemsetAsync(dinv, 0, N_NODES * sizeof(float), stream);
  hipMemsetAsync(stats, 0, 4096 * sizeof(float), stream);
  hipMemsetAsync(accum, 0, 256 * sizeof(float), stream);
  hipMemsetAsync(AGG, 0, (size_t)NROWS * COUT * sizeof(float), stream);

  prep_weights<<<64, 256, 0, stream>>>(gcn_w, tconv_w, res_w, Wp, Bwp, Brp);
  deg_kernel<<<(N_EDGES + 255) / 256, 256, 0, stream>>>(ei, dinv);
  dinv_kernel<<<(N_NODES + 255) / 256, 256, 0, stream>>>(dinv);

  // h = x @ gcn_w  (20000 row-tiles, 8 waves/block)
  gemm_gcn<<<2500, 256, 0, stream>>>(x, Wp, H);

  // scatter aggregation
  edge_agg<<<(int)(((long long)N_EDGES * T_DIM + 255) / 256), 256, 0, stream>>>(ei, dinv, H, AGG);
  self_agg<<<(NROWS + 255) / 256, 256, 0, stream>>>(dinv, H, AGG);

  // BN per timestep + ReLU (in place -> hs)
  bns_stats_k<<<dim3(T_DIM, 16), 256, 0, stream>>>(AGG, stats);
  bns_fin<<<8, 256, 0, stream>>>(stats, bns_g, bns_b);
  bns_apply<<<20000, 256, 0, stream>>>(AGG, stats);

  // temporal conv -> d_out (pre-BN)
  tconv_gemm<<<2500, 256, 0, stream>>>(AGG, Bwp, out);

  // global BN stats
  bnt_stats_k<<<512, 256, 0, stream>>>(out, accum);
  bnt_fin<<<1, 64, 0, stream>>>(accum, bnt_g, bnt_b);

  // fused BN+ReLU + residual WMMA GEMM + res_b, in place on d_out
  final_res<<<2500, 256, 0, stream>>>(x, Brp, accum, res_b, out);
}